// GCN_41094247088650
// MI455X (gfx1250) — compile-verified
//
#include <hip/hip_runtime.h>
#include <hip/hip_bf16.h>

typedef __attribute__((ext_vector_type(2))) float v2f;
typedef __attribute__((ext_vector_type(8))) float v8f;

#define NN 100000
#define EE 1600000
#define GG 500
#define NF 128
#define MD 30

// ---------------------------------------------------------------------------
// GEMM: H[nrows x 128] = X[nrows x 128] @ W[128 x 128] using V_WMMA_F32_16X16X4_F32
// Block = 256 threads = 8 waves; each wave owns one 16x16 output tile, the 8
// waves of a block cover one 16-row strip across all 128 output columns.
// ---------------------------------------------------------------------------
__global__ __launch_bounds__(256) void gemm_xw_wmma(const float* __restrict__ X,
                                                    const float* __restrict__ W,
                                                    float* __restrict__ Hout,
                                                    int nrows) {
  const int wave = threadIdx.x >> 5;   // 0..7 -> output column tile
  const int lane = threadIdx.x & 31;
  const int row0 = blockIdx.x * 16;
  if (row0 >= nrows) return;
  const int col0 = wave * 16;
  const int m    = lane & 15;          // row within tile (A) / col within tile (B,C)
  const int half = lane >> 4;          // selects K-pair (A,B) / M-half (C,D)

  v8f c = {0.f, 0.f, 0.f, 0.f, 0.f, 0.f, 0.f, 0.f};
  const float* xrow = X + (size_t)(row0 + m) * NF;

  for (int k0 = 0; k0 < NF; k0 += 4) {
    const int ka = k0 + 2 * half;
    v2f a, b;
    a.x = xrow[ka];
    a.y = xrow[ka + 1];
    b.x = W[(size_t)ka * NF + col0 + m];
    b.y = W[(size_t)(ka + 1) * NF + col0 + m];
    c = __builtin_amdgcn_wmma_f32_16x16x4_f32(false, a, false, b,
                                              (short)0, c, false, false);
  }
#pragma unroll
  for (int r = 0; r < 8; ++r) {
    const int mm = r + 8 * half;
    Hout[(size_t)(row0 + mm) * NF + col0 + m] = c[r];
  }
}

// ---------------------------------------------------------------------------
// Degree / normalization
// ---------------------------------------------------------------------------
__global__ void deg_init(float* __restrict__ deg, int n) {
  int i = blockIdx.x * blockDim.x + threadIdx.x;
  if (i < n) deg[i] = 1.0f;  // self-loop weight
}

__global__ void deg_edges(const int* __restrict__ dst, const float* __restrict__ ew,
                          float* __restrict__ deg, int e) {
  int i = blockIdx.x * blockDim.x + threadIdx.x;
  if (i < e) atomicAdd(&deg[dst[i]], ew[i]);
}

__global__ void make_dinv(const float* __restrict__ deg, float* __restrict__ dinv, int n) {
  int i = blockIdx.x * blockDim.x + threadIdx.x;
  if (i < n) {
    float d = deg[i];
    dinv[i] = (d > 0.0f) ? rsqrtf(d) : 0.0f;
  }
}

// ---------------------------------------------------------------------------
// Aggregation: agg = D^-1/2 (A+I) D^-1/2 h
// ---------------------------------------------------------------------------

// Self-loop init, float4-vectorized: agg = dinv[i]^2 * h  (n*NF/4 threads)
__global__ void self_init(const float4* __restrict__ h, const float* __restrict__ dinv,
                          float4* __restrict__ agg, int nvec) {
  int t = blockIdx.x * blockDim.x + threadIdx.x;
  if (t >= nvec) return;
  int i = t >> 5;                      // 32 float4 per node row
  float s = dinv[i];
  float s2 = s * s;
  float4 v = h[t];
  v.x *= s2; v.y *= s2; v.z *= s2; v.w *= s2;
  agg[t] = v;
}

// One wave (32 lanes) per edge; each lane owns 4 contiguous features:
// one b128 load of the message + 4 fp32 atomics into the destination row.
__global__ __launch_bounds__(256) void edge_scatter(const int* __restrict__ src,
                                                    const int* __restrict__ dst,
                                                    const float* __restrict__ ew,
                                                    const float* __restrict__ dinv,
                                                    const float4* __restrict__ h,
                                                    float* __restrict__ agg, int e) {
  const int wave = threadIdx.x >> 5;
  const int lane = threadIdx.x & 31;
  const int ei = blockIdx.x * 8 + wave;
  if (ei >= e) return;
  const int s = src[ei];
  const int d = dst[ei];
  const float coef = dinv[s] * ew[ei] * dinv[d];

  float4 v = h[(size_t)s * (NF / 4) + lane];
  float* dp = agg + (size_t)d * NF + lane * 4;
  atomicAdd(dp + 0, coef * v.x);
  atomicAdd(dp + 1, coef * v.y);
  atomicAdd(dp + 2, coef * v.z);
  atomicAdd(dp + 3, coef * v.w);
}

// Bias + ReLU, float4-vectorized (total = n*NF/4 threads)
__global__ void bias_relu(float4* __restrict__ agg, const float4* __restrict__ b,
                          int nvec) {
  int t = blockIdx.x * blockDim.x + threadIdx.x;
  if (t >= nvec) return;
  float4 bb = b[t & 31];
  float4 v = agg[t];
  v.x = fmaxf(v.x + bb.x, 0.0f);
  v.y = fmaxf(v.y + bb.y, 0.0f);
  v.z = fmaxf(v.z + bb.z, 0.0f);
  v.w = fmaxf(v.w + bb.w, 0.0f);
  agg[t] = v;
}

// ---------------------------------------------------------------------------
// Global mean pool + segment-min(first node)
// ---------------------------------------------------------------------------
__global__ void pool_init(float* __restrict__ pooled, float* __restrict__ cnt,
                          int* __restrict__ first) {
  int t = blockIdx.x * blockDim.x + threadIdx.x;
  if (t < GG * NF) pooled[t] = 0.0f;
  if (t < GG) { cnt[t] = 0.0f; first[t] = 0x7fffffff; }
}

__global__ void pool_accum(const float4* __restrict__ h, const int* __restrict__ batch,
                           float* __restrict__ pooled, float* __restrict__ cnt,
                           int* __restrict__ first, int nvec) {
  int t = blockIdx.x * blockDim.x + threadIdx.x;
  if (t >= nvec) return;
  int i  = t >> 5;                     // node index
  int f4 = t & 31;                     // float4 index within row
  int g  = batch[i];
  float4 v = h[t];
  float* pp = pooled + (size_t)g * NF + f4 * 4;
  atomicAdd(pp + 0, v.x);
  atomicAdd(pp + 1, v.y);
  atomicAdd(pp + 2, v.z);
  atomicAdd(pp + 3, v.w);
  if (f4 == 0) {
    atomicAdd(&cnt[g], 1.0f);
    atomicMin(&first[g], i);
  }
}

// ---------------------------------------------------------------------------
// Final: md = relu(metadata[idx] @ Wm + bm); out = [pooled_mean, md] @ Wf + bf
// One block (128 threads) per graph.
// ---------------------------------------------------------------------------
__global__ __launch_bounds__(128) void finalize(const float* __restrict__ pooled,
                                                const float* __restrict__ cnt,
                                                const int* __restrict__ first,
                                                const float* __restrict__ metadata,
                                                const float* __restrict__ Wm,
                                                const float* __restrict__ bm,
                                                const float* __restrict__ Wf,
                                                const float* __restrict__ bf,
                                                float* __restrict__ out) {
  __shared__ float red[128];
  const int g  = blockIdx.x;
  const int hh = threadIdx.x;

  float c     = cnt[g];
  float denom = c > 1.0f ? c : 1.0f;
  float pm    = pooled[(size_t)g * NF + hh] / denom;

  int idx = first[g] % GG;  // metadata.shape[0] == G
  float acc = bm[hh];
#pragma unroll
  for (int k = 0; k < MD; ++k)
    acc += metadata[idx * MD + k] * Wm[k * NF + hh];
  float md = acc > 0.0f ? acc : 0.0f;

  red[hh] = pm * Wf[hh] + md * Wf[NF + hh];
  __syncthreads();
  for (int s2 = 64; s2 > 0; s2 >>= 1) {
    if (hh < s2) red[hh] += red[hh + s2];
    __syncthreads();
  }
  if (hh == 0) out[g] = red[0] + bf[0];
}

// ---------------------------------------------------------------------------
extern "C" void kernel_launch(void* const* d_in, const int* in_sizes, int n_in,
                              void* d_out, int out_size, void* d_ws, size_t ws_size,
                              hipStream_t stream) {
  (void)in_sizes; (void)n_in; (void)out_size; (void)ws_size;

  const float* x        = (const float*)d_in[0];
  const int*   eidx     = (const int*)d_in[1];      // [2, E]: src then dst
  const float* eattr    = (const float*)d_in[2];
  const int*   batch    = (const int*)d_in[3];
  const float* metadata = (const float*)d_in[4];
  const float* W1       = (const float*)d_in[5];
  const float* b1       = (const float*)d_in[6];
  const float* W2       = (const float*)d_in[7];
  const float* b2       = (const float*)d_in[8];
  const float* Wm       = (const float*)d_in[9];
  const float* bm       = (const float*)d_in[10];
  const float* Wf       = (const float*)d_in[11];
  const float* bf       = (const float*)d_in[12];
  float*       out      = (float*)d_out;

  const int* src = eidx;
  const int* dst = eidx + EE;

  // Workspace carve-up
  float* bufA   = (float*)d_ws;                 // N*NF
  float* bufB   = bufA + (size_t)NN * NF;       // N*NF
  float* deg    = bufB + (size_t)NN * NF;       // N
  float* dinv   = deg + NN;                     // N
  float* pooled = dinv + NN;                    // G*NF
  float* cnt    = pooled + (size_t)GG * NF;     // G
  int*   first  = (int*)(cnt + GG);             // G

  const int NVEC        = NN * (NF / 4);             // 3.2M float4 elements
  const int nodeBlocks  = (NN + 255) / 256;
  const int edgeBlocks  = (EE + 255) / 256;
  const int vecBlocks   = (NVEC + 255) / 256;        // 12500
  const int edgeWaveBlk = (EE + 7) / 8;              // 1 wave per edge, 8/block
  const int gemmBlocks  = NN / 16;                   // 6250

  // Degree normalization (shared by both conv layers)
  deg_init<<<nodeBlocks, 256, 0, stream>>>(deg, NN);
  deg_edges<<<edgeBlocks, 256, 0, stream>>>(dst, eattr, deg, EE);
  make_dinv<<<nodeBlocks, 256, 0, stream>>>(deg, dinv, NN);

  // --- Layer 1: h1 = relu(norm-agg(x @ W1) + b1) ---
  gemm_xw_wmma<<<gemmBlocks, 256, 0, stream>>>(x, W1, bufA, NN);
  self_init<<<vecBlocks, 256, 0, stream>>>((const float4*)bufA, dinv,
                                           (float4*)bufB, NVEC);
  edge_scatter<<<edgeWaveBlk, 256, 0, stream>>>(src, dst, eattr, dinv,
                                                (const float4*)bufA, bufB, EE);
  bias_relu<<<vecBlocks, 256, 0, stream>>>((float4*)bufB, (const float4*)b1, NVEC);

  // --- Layer 2: h2 = relu(norm-agg(h1 @ W2) + b2) ---
  gemm_xw_wmma<<<gemmBlocks, 256, 0, stream>>>(bufB, W2, bufA, NN);
  self_init<<<vecBlocks, 256, 0, stream>>>((const float4*)bufA, dinv,
                                           (float4*)bufB, NVEC);
  edge_scatter<<<edgeWaveBlk, 256, 0, stream>>>(src, dst, eattr, dinv,
                                                (const float4*)bufA, bufB, EE);
  bias_relu<<<vecBlocks, 256, 0, stream>>>((float4*)bufB, (const float4*)b2, NVEC);

  // --- Pool + metadata + final dense ---
  pool_init<<<(GG * NF + 255) / 256, 256, 0, stream>>>(pooled, cnt, first);
  pool_accum<<<vecBlocks, 256, 0, stream>>>((const float4*)bufB, batch,
                                            pooled, cnt, first, NVEC);
  finalize<<<GG, 128, 0, stream>>>(pooled, cnt, first, metadata, Wm, bm, Wf, bf, out);
}